// Attention_23716809409022
// MI455X (gfx1250) — compile-verified
//
#include <hip/hip_runtime.h>

// Fused per-token rank-1 attention for MI455X (gfx1250, wave32).
//   q = xWq^T+bq ; k = (xWk^T+bk)/sqrt(D) ; v = xWv^T+bv
//   attn[n,a] = sum_b softmax_b(q[n,a]*k[n,b]) * v[n,b] ; out = attn Wo^T + bo
// All GEMM work uses V_WMMA_F32_16X16X4_F32; everything per 16-token tile
// stays in LDS (no HBM intermediates). Softmax row-max uses the rank-1
// closed form: max_b(t*k_b) = t>0 ? t*kmax : t*kmin.

#define DHID 128
#define LDP  132   // padded LDS row stride (floats): 132%64==4 -> conflict-free k-strided frags
#define TB   16    // tokens per block

typedef __attribute__((ext_vector_type(2))) float v2f;
typedef __attribute__((ext_vector_type(8))) float v8f;

__device__ __forceinline__ v8f wmma4(v2f a, v2f b, v8f c) {
  // 8 args: (neg_a, A, neg_b, B, c_mod, C, reuse_a, reuse_b)
  return __builtin_amdgcn_wmma_f32_16x16x4_f32(false, a, false, b, (short)0, c,
                                               false, false);
}

// A fragment (16x4 f32) from an LDS tile with row stride LDP:
// lane l holds row (l&15), K = k0 + 2*(l>>4) + {0,1}
__device__ __forceinline__ v2f fragA_lds(const float* base, int lane, int k0) {
  return *(const v2f*)(base + (lane & 15) * LDP + k0 + ((lane >> 4) << 1));
}
// B fragment (4x16 f32) for x@W^T: B[k][n] = W[n][k]  (W row-major DxD)
__device__ __forceinline__ v2f fragB_w(const float* __restrict__ W, int colb,
                                       int lane, int k0) {
  return *(const v2f*)(W + (colb + (lane & 15)) * DHID + k0 + ((lane >> 4) << 1));
}

__global__ __launch_bounds__(256)
void Attention_fused_kernel(const float* __restrict__ x,
                            const float* __restrict__ Wq, const float* __restrict__ bq,
                            const float* __restrict__ Wk, const float* __restrict__ bk,
                            const float* __restrict__ Wv, const float* __restrict__ bv,
                            const float* __restrict__ Wo, const float* __restrict__ bo,
                            float* __restrict__ out) {
  const int tid  = threadIdx.x;
  const int lane = tid & 31;
  const int wid  = tid >> 5;            // 0..7 (8 waves)
  const int n0   = blockIdx.x * TB;     // token base for this block
  const float SCALE = 0.0883883476483184f;  // 1/sqrt(128)

  __shared__ float xs [TB * LDP];
  __shared__ float qs [TB * LDP];
  __shared__ float ksh[TB * LDP];
  __shared__ float vsh[TB * LDP];
  __shared__ float ash[TB * LDP];
  __shared__ float pmax[TB][16], pmin[TB][16];
  __shared__ float kmaxs[TB], kmins[TB];

  // ---- Stage 0: coalesced stage of x tile into LDS -------------------------
#pragma unroll
  for (int i = 0; i < 8; ++i) {
    int idx = tid + i * 256;
    int r = idx >> 7, c = idx & 127;
    xs[r * LDP + c] = x[(n0 + r) * DHID + c];
  }
  __syncthreads();

  // ---- Stage 1: q,k,v projections. Wave w owns output cols [16w,16w+16). ---
  const int colb = wid * 16;
  v8f aq = {}, ak = {}, av = {};
#pragma unroll 8
  for (int kk = 0; kk < 32; ++kk) {
    const int k0 = kk * 4;
    v2f a  = fragA_lds(xs, lane, k0);        // shared A frag for all 3 GEMMs
    v2f bqf = fragB_w(Wq, colb, lane, k0);
    v2f bkf = fragB_w(Wk, colb, lane, k0);
    v2f bvf = fragB_w(Wv, colb, lane, k0);
    aq = wmma4(a, bqf, aq);
    ak = wmma4(a, bkf, ak);
    av = wmma4(a, bvf, av);
  }
  {
    const int col   = colb + (lane & 15);
    const int rbase = (lane >> 4) << 3;      // C vgpr r -> row r + 8*(lane>>4)
    const float bqv = bq[col], bkv = bk[col], bvv = bv[col];
#pragma unroll
    for (int r = 0; r < 8; ++r) {
      const int row = rbase + r;
      qs [row * LDP + col] = aq[r] + bqv;
      ksh[row * LDP + col] = (ak[r] + bkv) * SCALE;   // fold 1/sqrt(D) into k
      vsh[row * LDP + col] = av[r] + bvv;
    }
  }
  __syncthreads();

  // ---- Stage 2: per-token kmax/kmin (for closed-form softmax row-max) ------
  {
    const int token = tid >> 4, part = tid & 15;
    const float* kp = ksh + token * LDP + part * 8;
    float mx = kp[0], mn = kp[0];
#pragma unroll
    for (int i = 1; i < 8; ++i) { mx = fmaxf(mx, kp[i]); mn = fminf(mn, kp[i]); }
    pmax[token][part] = mx; pmin[token][part] = mn;
  }
  __syncthreads();
  if (tid < TB) {
    float mx = pmax[tid][0], mn = pmin[tid][0];
#pragma unroll
    for (int i = 1; i < 16; ++i) { mx = fmaxf(mx, pmax[tid][i]); mn = fminf(mn, pmin[tid][i]); }
    kmaxs[tid] = mx; kmins[tid] = mn;
  }
  __syncthreads();

  // ---- Stage 3: softmax-weighted sum. 8 (n,a) pairs per thread. ------------
#pragma unroll
  for (int p = 0; p < 8; ++p) {
    const int pair = tid + p * 256;
    const int n = pair >> 7, a = pair & 127;
    const float t = qs[n * LDP + a];
    const float m = (t > 0.f) ? t * kmaxs[n] : t * kmins[n];  // exact row max
    const float* krow = ksh + n * LDP;
    const float* vrow = vsh + n * LDP;
    float s = 0.f, acc = 0.f;
#pragma unroll 4
    for (int b = 0; b < DHID; ++b) {
      const float e = __expf(fmaf(t, krow[b], -m));
      s += e;
      acc = fmaf(e, vrow[b], acc);
    }
    ash[n * LDP + a] = acc / s;
  }
  __syncthreads();

  // ---- Stage 4: output projection, tile per wave, store to HBM -------------
  v8f co = {};
#pragma unroll 8
  for (int kk = 0; kk < 32; ++kk) {
    const int k0 = kk * 4;
    v2f a = fragA_lds(ash, lane, k0);
    v2f b = fragB_w(Wo, colb, lane, k0);
    co = wmma4(a, b, co);
  }
  {
    const int col   = colb + (lane & 15);
    const int rbase = (lane >> 4) << 3;
    const float bov = bo[col];
#pragma unroll
    for (int r = 0; r < 8; ++r) {
      const int row = rbase + r;
      out[(n0 + row) * DHID + col] = co[r] + bov;
    }
  }
}

extern "C" void kernel_launch(void* const* d_in, const int* in_sizes, int n_in,
                              void* d_out, int out_size, void* d_ws, size_t ws_size,
                              hipStream_t stream) {
  const float* x  = (const float*)d_in[0];
  const float* Wq = (const float*)d_in[1];
  const float* bq = (const float*)d_in[2];
  const float* Wk = (const float*)d_in[3];
  const float* bk = (const float*)d_in[4];
  const float* Wv = (const float*)d_in[5];
  const float* bv = (const float*)d_in[6];
  const float* Wo = (const float*)d_in[7];
  const float* bo = (const float*)d_in[8];
  float* out = (float*)d_out;

  const int nTok   = in_sizes[0] / DHID;   // 8192
  const int blocks = nTok / TB;            // 512
  Attention_fused_kernel<<<blocks, 256, 0, stream>>>(x, Wq, bq, Wk, bk,
                                                     Wv, bv, Wo, bo, out);
}